// ModelNew_80908593923132
// MI455X (gfx1250) — compile-verified
//
#include <hip/hip_runtime.h>
#include <hip/hip_bf16.h>

typedef float v2f __attribute__((ext_vector_type(2)));
typedef float v8f __attribute__((ext_vector_type(8)));

#define CIN   3
#define DDIM  16
#define HDIM  128
#define WDIM  128
#define COUT  24
#define DPRIME 14
#define HP    126
#define WP    126
#define KTOT  81      // CIN * 27
#define NCHUNK 21     // K padded to 84, as 42 pairs / 21 chunks

#define WS 66                      // LDS width: 64 cols + 2 halo
#define DSTRIDE (3*WS*4)           // 792 B between consecutive depths
#define XS_SIZE (CIN*DDIM*3*WS)    // 9504 floats = 38016 B

// ---- compile-time K-permutation tables: 42 pairs, 2 delta classes ----
// pairs 0..29  : delta 1  (chunks 0..14)
//   p = ci*9+t (t=kd*3+kh): elements (ci,kd,kh,kw0),(ci,kd,kh,kw1)
//   p = 27+ci           : pad pair, slot0 = zero-weight, slot1 = (ci,2,2,2)
// pairs 30..41 : delta 66 (chunks 15..20)
//   kw=2 chain j -> offset 66*j+2; pairs (2q,2q+1), q=0..3 per ci
struct KTables { int pa0[42]; int pa1[42]; int boff[42]; };
static constexpr KTables makeTables() {
    KTables t{};
    int p = 0;
    for (int ci = 0; ci < 3; ++ci)
        for (int tt = 0; tt < 9; ++tt) {
            const int kd = tt / 3, kh = tt % 3;
            t.pa0[p]  = ci * 27 + kd * 9 + kh * 3 + 0;
            t.pa1[p]  = ci * 27 + kd * 9 + kh * 3 + 1;
            t.boff[p] = ci * 3168 + kd * 198 + kh * 66 + 0;
            ++p;
        }
    for (int ci = 0; ci < 3; ++ci) {              // leftover (ci,2,2,2) + pad
        t.pa0[p]  = -1;                           // slot0: zero weight (loads kw=1 value)
        t.pa1[p]  = ci * 27 + 26;                 // slot1: real element (ci,2,2,2)
        t.boff[p] = ci * 3168 + 2 * 198 + 2 * 66 + 1;   // base = element - 1
        ++p;
    }
    for (int ci = 0; ci < 3; ++ci)
        for (int q = 0; q < 4; ++q) {
            const int j0 = 2 * q, j1 = 2 * q + 1;
            t.pa0[p]  = ci * 27 + (j0 / 3) * 9 + (j0 % 3) * 3 + 2;
            t.pa1[p]  = ci * 27 + (j1 / 3) * 9 + (j1 % 3) * 3 + 2;
            t.boff[p] = ci * 3168 + (j0 / 3) * 198 + (j0 % 3) * 66 + 2;
            ++p;
        }
    return t;
}
static constexpr KTables TB = makeTables();

__global__ __launch_bounds__(128) void conv3d_min_softmax_wmma(
    const float* __restrict__ x,
    const float* __restrict__ w,
    const float* __restrict__ bias,
    float* __restrict__ out)
{
    __shared__ float xs[XS_SIZE];

    const int b   = blockIdx.z;
    const int h0  = blockIdx.y;          // h' in [0,126)
    const int wb  = blockIdx.x * 64;     // 0 or 64
    const int tid = threadIdx.x;
    const int lane = tid & 31;
    const int wave = tid >> 5;

    // ---- cooperative stage of x slab into LDS: [ci][d][hh][ww] ----
    for (int i = tid; i < XS_SIZE; i += 128) {
        int ww = i % WS;
        int r  = i / WS;
        int hh = r % 3;  r /= 3;
        int d  = r % DDIM;
        int ci = r / DDIM;
        int gw = wb + ww; gw = (gw > WDIM - 1) ? (WDIM - 1) : gw;  // clamp halo (feeds only invalid cols)
        int gh = h0 + hh;                                           // <= 127 always
        xs[i] = x[(((b * CIN + ci) * DDIM + d) * HDIM + gh) * WDIM + gw];
    }
    __syncthreads();

    const int  m  = lane & 15;            // A: M row / B: N col
    const bool lo = (lane < 16);          // K-slot pair 0 vs pair 1 of each chunk
    const int  c0 = wave * 16;            // this wave's LDS column base

    // ---- preload A fragments (weights, permuted) + per-chunk B base offsets ----
    v2f a0[NCHUNK], a1[NCHUNK];
    int bo[NCHUNK];                       // byte offsets, advanced by DSTRIDE per depth
#pragma unroll
    for (int kc = 0; kc < NCHUNK; ++kc) {
        const int ka0 = lo ? TB.pa0[2 * kc] : TB.pa0[2 * kc + 1];   // select of literals
        const int ka1 = lo ? TB.pa1[2 * kc] : TB.pa1[2 * kc + 1];
        const int bof = lo ? TB.boff[2 * kc] : TB.boff[2 * kc + 1];
        a0[kc].x = (ka0 >= 0) ? w[m * KTOT + ka0] : 0.f;
        a0[kc].y = (ka1 >= 0) ? w[m * KTOT + ka1] : 0.f;
        const int c1 = m + 16;            // channels 16..31 (24..31 zero-padded)
        a1[kc].x = (ka0 >= 0 && c1 < COUT) ? w[c1 * KTOT + ka0] : 0.f;
        a1[kc].y = (ka1 >= 0 && c1 < COUT) ? w[c1 * KTOT + ka1] : 0.f;
        bo[kc] = 4 * (bof + c0 + m);
    }

    const char* xsb = (const char*)xs;

    v8f mn0, mn1;
#pragma unroll
    for (int r = 0; r < 8; ++r) { mn0[r] = 3.402823466e+38f; mn1[r] = 3.402823466e+38f; }

    // ---- depth loop: per iteration 21 ds_load_2addr_b32 (pair = ready-made
    //      B operand), 21 base increments, 42 WMMAs, running min ----
#pragma unroll 1
    for (int dp = 0; dp < DPRIME; ++dp) {
        v8f acc0 = {}, acc1 = {};
#pragma unroll
        for (int kc = 0; kc < NCHUNK; ++kc) {
            const int o = bo[kc];
            v2f bf;
            bf.x = *(const float*)(xsb + o);
            bf.y = *(const float*)(xsb + o + ((kc < 15) ? 4 : 264));  // delta 1 or 66 dwords
            bo[kc] = o + DSTRIDE;
            acc0 = __builtin_amdgcn_wmma_f32_16x16x4_f32(
                false, a0[kc], false, bf, (short)0, acc0, false, false);
            acc1 = __builtin_amdgcn_wmma_f32_16x16x4_f32(
                false, a1[kc], false, bf, (short)0, acc1, false, false);
        }
#pragma unroll
        for (int r = 0; r < 8; ++r) {
            mn0[r] = fminf(mn0[r], acc0[r]);
            mn1[r] = fminf(mn1[r], acc1[r]);
        }
    }

    // ---- gather 24 channels into lanes 0..15, bias + softmax, store ----
    // C/D layout: VGPR r holds M=r (lanes 0-15) and M=r+8 (lanes 16-31).
    float v[COUT];
#pragma unroll
    for (int r = 0; r < 8; ++r) {
        float hi = __shfl_xor(mn0[r], 16, 32);  // lanes<16 receive channels 8..15
        v[r]      = mn0[r];
        v[8 + r]  = hi;
        v[16 + r] = mn1[r];                     // lanes<16 hold channels 16..23
    }

    if (lane < 16) {
        const int wo = wb + c0 + lane;          // global w'
        if (wo < WP) {
            float mx = -3.402823466e+38f;
#pragma unroll
            for (int c = 0; c < COUT; ++c) { v[c] += bias[c]; mx = fmaxf(mx, v[c]); }
            float sum = 0.f;
#pragma unroll
            for (int c = 0; c < COUT; ++c) { v[c] = __expf(v[c] - mx); sum += v[c]; }
            const float inv = 1.0f / sum;
#pragma unroll
            for (int c = 0; c < COUT; ++c) {
                out[((b * COUT + c) * HP + h0) * WP + wo] = v[c] * inv;
            }
        }
    }
}

extern "C" void kernel_launch(void* const* d_in, const int* in_sizes, int n_in,
                              void* d_out, int out_size, void* d_ws, size_t ws_size,
                              hipStream_t stream) {
    (void)in_sizes; (void)n_in; (void)d_ws; (void)ws_size; (void)out_size;
    const float* x    = (const float*)d_in[0];
    const float* wgt  = (const float*)d_in[1];
    const float* bias = (const float*)d_in[2];
    float* out = (float*)d_out;

    dim3 grid(2, HP, 16);   // (w half, h', batch)
    dim3 block(128);        // 4 wave32s, one 16-wide N tile each
    conv3d_min_softmax_wmma<<<grid, block, 0, stream>>>(x, wgt, bias, out);
}